// LSTM_13967233647581
// MI455X (gfx1250) — compile-verified
//
#include <hip/hip_runtime.h>

typedef __attribute__((ext_vector_type(2))) float v2f;
typedef __attribute__((ext_vector_type(8))) float v8f;

#define T_SEQ  131072
#define IN_DIM 30
#define H_DIM  100
#define G4     400      // 4*H

// ---------------------------------------------------------------------------
// Kernel 1: x_gates[T,400] = input_seq[T,30] @ W_ih.T[30,400] + (b_ih + b_hh)
// f32 WMMA 16x16x4, one 16x16 output tile per wave. K=30 padded to 32 with
// zeros (lanes whose K index >= 30 contribute 0). Memory-bound: ~226 MB total
// traffic -> ~10us at 23.3 TB/s.
// ---------------------------------------------------------------------------
__global__ __launch_bounds__(256) void lstm_xgates_wmma(
    const float* __restrict__ x,    // [T, 30]
    const float* __restrict__ Wih,  // [400, 30] row-major
    const float* __restrict__ bih,  // [400]
    const float* __restrict__ bhh,  // [400]
    float* __restrict__ xg)         // [T, 400]
{
    const int wave = blockIdx.x * (blockDim.x >> 5) + (threadIdx.x >> 5);
    const int lane = threadIdx.x & 31;
    const int NT   = G4 / 16;          // 25 tiles along N
    const int mt   = wave / NT;        // tile along T
    const int nt   = wave % NT;        // tile along 4H
    const int m0   = mt * 16;
    const int n0   = nt * 16;
    const int row  = lane & 15;        // N (for B,C,D) / M (for A)
    const int grp  = lane >> 4;        // half-wave group

    // C init: bias depends only on N = n0 + row (same for all 8 C VGPRs)
    const float bias = bih[n0 + row] + bhh[n0 + row];
    v8f c;
#pragma unroll
    for (int v = 0; v < 8; ++v) c[v] = bias;

#pragma unroll
    for (int kk = 0; kk < 32; kk += 4) {
        const int k0 = kk + 2 * grp;   // K index held by a[0]/b[0] in this lane
        v2f a, b;
        // A[M=row, K=k0..k0+1] = x[m0+row, k]
        a.x = (k0     < IN_DIM) ? x[(m0 + row) * IN_DIM + k0]     : 0.0f;
        a.y = (k0 + 1 < IN_DIM) ? x[(m0 + row) * IN_DIM + k0 + 1] : 0.0f;
        // B[K=k0..k0+1, N=row] = W_ih.T[k, n0+row] = W_ih[n0+row, k]
        b.x = (k0     < IN_DIM) ? Wih[(n0 + row) * IN_DIM + k0]     : 0.0f;
        b.y = (k0 + 1 < IN_DIM) ? Wih[(n0 + row) * IN_DIM + k0 + 1] : 0.0f;
        c = __builtin_amdgcn_wmma_f32_16x16x4_f32(
                /*neg_a=*/false, a, /*neg_b=*/false, b,
                /*c_mod=*/(short)0, c, /*reuse_a=*/false, /*reuse_b=*/false);
    }

    // D element (M = v + 8*grp, N = row)
#pragma unroll
    for (int v = 0; v < 8; ++v)
        xg[(size_t)(m0 + v + 8 * grp) * G4 + n0 + row] = c[v];
}

// ---------------------------------------------------------------------------
// Kernel 2: sequential LSTM scan. One workgroup, 416 threads (13 waves).
// Thread r < 400 owns gate row r: W_hh row held in 100 VGPRs, h broadcast
// from LDS. Threads 0..99 own c[j] in a register. xg rows stream from L2
// (kernel 1 just wrote them; 200 MB vs 192 MB L2), with prefetch lookahead.
// ---------------------------------------------------------------------------
__global__ __launch_bounds__(416) void lstm_scan(
    const float* __restrict__ xg,    // [T, 400]
    const float* __restrict__ Whh,   // [400, 100] row-major
    const float* __restrict__ Wlin,  // [1, 100]
    const float* __restrict__ blin,  // [1]
    float* __restrict__ out)         // [1]
{
    __shared__ float h_s[H_DIM];
    __shared__ float act_s[G4];
    __shared__ float red_s[H_DIM];

    const int r = threadIdx.x;

    // Preload this thread's W_hh row into registers (100 VGPRs)
    float w[H_DIM];
    if (r < G4) {
#pragma unroll 4
        for (int k = 0; k < H_DIM; ++k) w[k] = Whh[r * H_DIM + k];
    }
    if (r < H_DIM) h_s[r] = 0.0f;
    float c = 0.0f;
    __syncthreads();

    for (int t = 0; t < T_SEQ; ++t) {
        if (r < G4) {
            if (t + 8 < T_SEQ)
                __builtin_prefetch(&xg[(size_t)(t + 8) * G4 + r], 0, 1);

            float acc = xg[(size_t)t * G4 + r];
#pragma unroll 4
            for (int k = 0; k < H_DIM; ++k) acc += w[k] * h_s[k];

            float g;
            if (r < 2 * H_DIM)      g = 1.0f / (1.0f + __expf(-acc)); // i, f
            else if (r < 3 * H_DIM) g = tanhf(acc);                   // g
            else                    g = 1.0f / (1.0f + __expf(-acc)); // o
            act_s[r] = g;
        }
        __syncthreads();
        if (r < H_DIM) {
            c = act_s[H_DIM + r] * c + act_s[r] * act_s[2 * H_DIM + r];
            h_s[r] = act_s[3 * H_DIM + r] * tanhf(c);
        }
        __syncthreads();
    }

    // Linear head on final h
    if (r < H_DIM) red_s[r] = h_s[r] * Wlin[r];
    __syncthreads();
    if (r == 0) {
        float s = blin[0];
        for (int k = 0; k < H_DIM; ++k) s += red_s[k];
        out[0] = s;
    }
}

// ---------------------------------------------------------------------------
extern "C" void kernel_launch(void* const* d_in, const int* in_sizes, int n_in,
                              void* d_out, int out_size, void* d_ws, size_t ws_size,
                              hipStream_t stream) {
    const float* x    = (const float*)d_in[0];  // input_seq [T,30]
    const float* Wih  = (const float*)d_in[1];  // [400,30]
    const float* Whh  = (const float*)d_in[2];  // [400,100]
    const float* bih  = (const float*)d_in[3];  // [400]
    const float* bhh  = (const float*)d_in[4];  // [400]
    const float* Wlin = (const float*)d_in[5];  // [1,100]
    const float* blin = (const float*)d_in[6];  // [1]
    float* out = (float*)d_out;
    float* xg  = (float*)d_ws;                  // needs T*400*4 = 200 MB scratch

    const int total_waves     = (T_SEQ / 16) * (G4 / 16);  // 204800 tiles
    const int waves_per_block = 256 / 32;                  // 8
    const int blocks          = total_waves / waves_per_block;

    lstm_xgates_wmma<<<blocks, 256, 0, stream>>>(x, Wih, bih, bhh, xg);
    lstm_scan<<<1, 416, 0, stream>>>(xg, Whh, Wlin, blin, out);
}